// Multi_Scale_Region_Distillation_Loss_3_33414845562966
// MI455X (gfx1250) — compile-verified
//
#include <hip/hip_runtime.h>
#include <math.h>

typedef float v2f __attribute__((ext_vector_type(2)));
typedef float v8f __attribute__((ext_vector_type(8)));

#define NCLS    32            // padded class slots (num_class = 21 at runtime)
#define THREADS 256           // 8 wave32s per block
#define VEC     2             // pixels per lane -> 64-bit coalesced loads
#define PIXB    (THREADS * VEC)
#define SRC     512           // pseudo-label map is (B,1,512,512)

struct Desc {
    const float* fn;
    const float* fo;
    int C, H, W, HW, tile0;
};

// ---------------------------------------------------------------------------
// init: zero the per-scale per-class accumulators and the output
// ---------------------------------------------------------------------------
__global__ __launch_bounds__(THREADS) void msrd_zero(float* __restrict__ acc,
                                                     float* __restrict__ out) {
    int t = blockIdx.x * blockDim.x + threadIdx.x;
    if (t < 4 * 2 * NCLS) acc[t] = 0.0f;
    if (t == 0) out[0] = 0.0f;
}

// ---------------------------------------------------------------------------
// main streaming kernel: one pass over all 4 scales, online softmax-KL per
// pixel, per-class (sum,count) accumulation via LDS then global atomics.
// ---------------------------------------------------------------------------
__global__ __launch_bounds__(THREADS)
void msrd_kl(Desc d0, Desc d1, Desc d2, Desc d3,
             const int* __restrict__ pl, const int* __restrict__ pnc,
             float* __restrict__ acc)
{
    __shared__ float lsum[NCLS];
    __shared__ float lcnt[NCLS];
    const int tid = threadIdx.x;
    if (tid < NCLS) { lsum[tid] = 0.0f; lcnt[tid] = 0.0f; }
    __syncthreads();

    // block -> scale
    Desc d; int scale;
    const int blk = blockIdx.x;
    if      (blk >= d3.tile0) { d = d3; scale = 3; }
    else if (blk >= d2.tile0) { d = d2; scale = 2; }
    else if (blk >= d1.tile0) { d = d1; scale = 1; }
    else                      { d = d0; scale = 0; }

    const int nc = pnc[0];
    const int HW = d.HW;
    const int p0 = (blk - d.tile0) * PIXB + tid * VEC;   // even; pair stays in one image
    const int b    = p0 / HW;
    const int sidx = p0 - b * HW;

    const size_t base = (size_t)b * d.C * HW + sidx;
    const float* __restrict__ pn = d.fn + base;
    const float* __restrict__ po = d.fo + base;

    const float NINF = -__builtin_inff();
    float mx0 = NINF, mx1 = NINF, my0 = NINF, my1 = NINF;
    float Sx0 = 0.f, Sx1 = 0.f, Sw0 = 0.f, Sw1 = 0.f, Sy0 = 0.f, Sy1 = 0.f;

    for (int c = 0; c < d.C; ++c) {
        const size_t off = (size_t)c * HW;
        const v2f xv = __builtin_nontemporal_load((const v2f*)(pn + off));
        const v2f yv = __builtin_nontemporal_load((const v2f*)(po + off));
        const float x0 = xv.x, x1 = xv.y, y0 = yv.x, y1 = yv.y;

        // pixel 0: online lse(x), Σ e^x (x-y), online lse(y)
        if (x0 > mx0) { float cr = __expf(mx0 - x0); Sx0 *= cr; Sw0 *= cr; mx0 = x0; }
        { float e = __expf(x0 - mx0); Sx0 += e; Sw0 += e * (x0 - y0); }
        if (y0 > my0) { Sy0 *= __expf(my0 - y0); my0 = y0; }
        Sy0 += __expf(y0 - my0);

        // pixel 1
        if (x1 > mx1) { float cr = __expf(mx1 - x1); Sx1 *= cr; Sw1 *= cr; mx1 = x1; }
        { float e = __expf(x1 - mx1); Sx1 += e; Sw1 += e * (x1 - y1); }
        if (y1 > my1) { Sy1 *= __expf(my1 - y1); my1 = y1; }
        Sy1 += __expf(y1 - my1);
    }

    // KL = Σp(x-y) - lse_x + lse_y
    const float kl0 = Sw0 / Sx0 - (mx0 + __logf(Sx0)) + (my0 + __logf(Sy0));
    const float kl1 = Sw1 / Sx1 - (mx1 + __logf(Sx1)) + (my1 + __logf(Sy1));

    // nearest-neighbor labels from the 512x512 map
    const int h0 = sidx / d.W, w0 = sidx - h0 * d.W;
    const int s1 = sidx + 1;
    const int h1 = s1 / d.W,   w1 = s1 - h1 * d.W;
    const int ih0 = (h0 * SRC) / d.H, iw0 = (w0 * SRC) / d.W;
    const int ih1 = (h1 * SRC) / d.H, iw1 = (w1 * SRC) / d.W;
    const int lab0 = pl[((size_t)b * SRC + ih0) * SRC + iw0];
    const int lab1 = pl[((size_t)b * SRC + ih1) * SRC + iw1];

    if ((unsigned)lab0 < (unsigned)nc) { atomicAdd(&lsum[lab0], kl0); atomicAdd(&lcnt[lab0], 1.0f); }
    if ((unsigned)lab1 < (unsigned)nc) { atomicAdd(&lsum[lab1], kl1); atomicAdd(&lcnt[lab1], 1.0f); }

    __syncthreads();
    if (tid < NCLS) {
        const float s = lsum[tid], c = lcnt[tid];
        if (s != 0.0f || c != 0.0f) {
            atomicAdd(&acc[scale * 2 * NCLS + tid], s);
            atomicAdd(&acc[scale * 2 * NCLS + NCLS + tid], c);
        }
    }
}

// ---------------------------------------------------------------------------
// epilogue: loss = w^T (coef ⊙ mean) summed over classes, as a 16x4 x 4x16
// WMMA (V_WMMA_F32_16X16X4_F32). One wave, EXEC all-1.
// A (16x4): every row = scale weights (1,2,3,4); K layout V0=K0|K2, V1=K1|K3.
// B (4x16): row K = scale, col N = class; V0 = K0|K2, V1 = K1|K3 (mirrored).
// Two column tiles cover classes 0..31; sum D row 0 over N.
// ---------------------------------------------------------------------------
__device__ __forceinline__ float msrd_val(const float* __restrict__ acc,
                                          int s, int cls, int nc, int noc) {
    if (cls >= nc) return 0.0f;
    const float sum = acc[s * 2 * NCLS + cls];
    const float cnt = acc[s * 2 * NCLS + NCLS + cls];
    const float mean = (cnt > 0.0f) ? (sum / cnt) : 0.0f;
    const float coef = (cls == 0) ? ((float)noc / (float)nc)
                                  : ((cls <= noc) ? 1.0f : 0.0f);
    return coef * mean;
}

__global__ __launch_bounds__(32)
void msrd_final(const float* __restrict__ acc,
                const int* __restrict__ pnc, const int* __restrict__ pnoc,
                float* __restrict__ out)
{
    const int lane = threadIdx.x;
    const int nc  = pnc[0];
    const int noc = pnoc[0];

    const int n    = lane & 15;     // column (class within tile)
    const int half = lane >> 4;     // lane half selects K pair {0,1} vs {2,3}

    // A: all 16 rows identical = scale weights; per-lane K mapping
    v2f a;
    a.x = half ? 3.0f : 1.0f;       // V0 -> K2 | K0
    a.y = half ? 4.0f : 2.0f;       // V1 -> K3 | K1

    // B tiles: classes 0..15 and 16..31
    v2f b0, b1;
    b0.x = msrd_val(acc, 2 * half + 0, n,      nc, noc);
    b0.y = msrd_val(acc, 2 * half + 1, n,      nc, noc);
    b1.x = msrd_val(acc, 2 * half + 0, 16 + n, nc, noc);
    b1.y = msrd_val(acc, 2 * half + 1, 16 + n, nc, noc);

    v8f c = {};
    v8f dacc = __builtin_amdgcn_wmma_f32_16x16x4_f32(
        false, a, false, b0, (short)0, c, false, false);
    dacc = __builtin_amdgcn_wmma_f32_16x16x4_f32(
        false, a, false, b1, (short)0, dacc, false, false);

    // row m=0 lives in VGPR0 on lanes 0..15 (lanes 16..31 hold row m=8, a duplicate)
    float v = (lane < 16) ? dacc[0] : 0.0f;
    for (int off = 16; off >= 1; off >>= 1) v += __shfl_down(v, off, 32);
    if (lane == 0) out[0] = v;
}

// ---------------------------------------------------------------------------
extern "C" void kernel_launch(void* const* d_in, const int* in_sizes, int n_in,
                              void* d_out, int out_size, void* d_ws, size_t ws_size,
                              hipStream_t stream) {
    (void)in_sizes; (void)n_in; (void)out_size; (void)ws_size;
    // setup_inputs() dict order: pl, fn0, fo0, fn1, fo1, fn2, fo2, fn3, fo3, nc, noc
    const int*   pl  = (const int*)d_in[0];
    const float* fn0 = (const float*)d_in[1];
    const float* fo0 = (const float*)d_in[2];
    const float* fn1 = (const float*)d_in[3];
    const float* fo1 = (const float*)d_in[4];
    const float* fn2 = (const float*)d_in[5];
    const float* fo2 = (const float*)d_in[6];
    const float* fn3 = (const float*)d_in[7];
    const float* fo3 = (const float*)d_in[8];
    const int*   pnc  = (const int*)d_in[9];
    const int*   pnoc = (const int*)d_in[10];

    float* out = (float*)d_out;
    float* acc = (float*)d_ws;   // 4 scales * 2 (sum,cnt) * NCLS floats = 1 KB

    const int B = 8;
    Desc D[4] = {
        { fn0, fo0,  256, 128, 128, 128 * 128, 0 },
        { fn1, fo1,  512,  64,  64,  64 *  64, 0 },
        { fn2, fo2, 1024,  32,  32,  32 *  32, 0 },
        { fn3, fo3, 2048,  16,  16,  16 *  16, 0 },
    };
    int tiles = 0;
    for (int i = 0; i < 4; ++i) {
        D[i].tile0 = tiles;
        tiles += (B * D[i].HW) / PIXB;   // all exact multiples of 512
    }

    msrd_zero<<<1, THREADS, 0, stream>>>(acc, out);
    msrd_kl<<<tiles, THREADS, 0, stream>>>(D[0], D[1], D[2], D[3], pl, pnc, acc);
    msrd_final<<<1, 32, 0, stream>>>(acc, pnc, pnoc, out);
}